// IAF_83571473646127
// MI455X (gfx1250) — compile-verified
//
#include <hip/hip_runtime.h>
#include <hip/hip_bf16.h>
#include <math.h>

// IAF / MADE inverse pass, B=1024, D=128, H=512.
// Incremental algorithm: maintain hidden pre-activations a[b,h] in LDS,
// per step d: sigma/mu via WMMA f32 16x16x4 (16 rows x H) . (H x 2 masked W2 cols),
// then rank-1 update a += z_d (x) W1m[:,d].

#define Bn   1024
#define Dn   128
#define Hn   512
#define ROWS 16          // batch rows per workgroup
#define NTHR 256         // 8 wave32
#define APAD 516         // padded row stride for a_pre (conflict-free WMMA A loads)

typedef __attribute__((ext_vector_type(2))) float v2f;
typedef __attribute__((ext_vector_type(8))) float v8f;

__global__ __launch_bounds__(NTHR)
void iaf_made_inverse(const float* __restrict__ x,
                      const float* __restrict__ W1,
                      const float* __restrict__ b1,
                      const float* __restrict__ W2,
                      const float* __restrict__ b2,
                      float* __restrict__ out) {
    __shared__ float a_pre[ROWS * APAD];   // hidden pre-activations (no bias)
    __shared__ float zsh[ROWS * Dn];       // produced z tile
    __shared__ float xsh[ROWS * Dn];       // input x tile
    __shared__ float b1sh[Hn];
    __shared__ float w2sig[Hn];            // masked W2 row d
    __shared__ float w2mu[Hn];             // masked W2 row D+d
    __shared__ float w1col[Hn];            // masked W1 column d
    __shared__ float sigred[ROWS];
    __shared__ float mured[ROWS];
    __shared__ float znew[ROWS];

    const int tid  = threadIdx.x;
    const int row0 = blockIdx.x * ROWS;

    // ---- one-time init ----
    for (int i = tid; i < ROWS * APAD; i += NTHR) a_pre[i] = 0.0f;
    for (int i = tid; i < Hn; i += NTHR)          b1sh[i]  = b1[i];
    for (int i = tid; i < ROWS * Dn; i += NTHR)   xsh[i]   = x[row0 * Dn + i];

    // lane-constant WMMA geometry (hoisted out of the d-loop)
    const int lane  = tid & 31;
    const int wave  = tid >> 5;          // 0..7 -> K slice of 64
    const int m     = lane & 15;         // A row / B column index
    const int half  = lane >> 4;         // 0/1
    const int kbase = wave * 64;
    const float* bsrc   = (m == 0) ? w2sig : w2mu;   // B column 0 = sigma, 1 = mu
    const float  bscale = (m < 2) ? 1.0f : 0.0f;     // columns >=2 are zero

    float lgd = 0.0f;                    // log-det accumulator (valid for tid<16)

    __syncthreads();

    for (int d = 0; d < Dn; ++d) {
        // ---- phase A: zero reductions, stage masked weights ----
        if (tid < ROWS) { sigred[tid] = 0.0f; mured[tid] = 0.0f; }
        for (int i = tid; i < Hn; i += NTHR) {      // 2 iterations
            unsigned hm   = (unsigned)i % 127u;     // m_hid - 1
            float msk2    = (hm < (unsigned)d) ? 1.0f : 0.0f;  // M2: m_out > m_hid
            float msk1    = (hm >= (unsigned)d) ? 1.0f : 0.0f; // M1: m_hid >= m_in
            w2sig[i] = W2[d * Hn + i] * msk2;
            w2mu[i]  = W2[(Dn + d) * Hn + i] * msk2;
            w1col[i] = W1[i * Dn + d] * msk1;
        }
        if (d + 1 < Dn && tid < 2) {
            // warm L2/L0 for next step's W2 rows (global_prefetch_b8)
            __builtin_prefetch(&W2[(Dn * tid + d + 1) * Hn], 0, 1);
        }
        __syncthreads();

        // ---- phase B: sigma/mu via f32 WMMA, K split across 8 waves ----
        v8f c0 = {}; v8f c1 = {};
        #pragma unroll
        for (int it = 0; it < 16; it += 2) {
            const int klo0 = kbase + it * 4 + 2 * half;
            const int klo1 = klo0 + 4;
            v2f a0, a1, bf0, bf1;
            {
                const int ab = m * APAD + klo0;
                a0.x = fmaxf(a_pre[ab]     + b1sh[klo0],     0.0f);
                a0.y = fmaxf(a_pre[ab + 1] + b1sh[klo0 + 1], 0.0f);
                bf0.x = bsrc[klo0]     * bscale;
                bf0.y = bsrc[klo0 + 1] * bscale;
            }
            {
                const int ab = m * APAD + klo1;
                a1.x = fmaxf(a_pre[ab]     + b1sh[klo1],     0.0f);
                a1.y = fmaxf(a_pre[ab + 1] + b1sh[klo1 + 1], 0.0f);
                bf1.x = bsrc[klo1]     * bscale;
                bf1.y = bsrc[klo1 + 1] * bscale;
            }
            c0 = __builtin_amdgcn_wmma_f32_16x16x4_f32(false, a0, false, bf0,
                                                       (short)0, c0, false, false);
            c1 = __builtin_amdgcn_wmma_f32_16x16x4_f32(false, a1, false, bf1,
                                                       (short)0, c1, false, false);
        }
        v8f cs = c0 + c1;

        // columns 0 (sigma) / 1 (mu); vgpr r holds row r + 8*half
        if (m < 2) {
            float* dst = (m == 0) ? sigred : mured;
            const int ro = half * 8;
            #pragma unroll
            for (int r = 0; r < 8; ++r) atomicAdd(&dst[ro + r], cs[r]);
        }
        __syncthreads();

        // ---- phase C: z[:,d], log_det update ----
        if (tid < ROWS) {
            float s  = sigred[tid] + b2[d];
            float mm = mured[tid] + b2[Dn + d];
            float zn = (xsh[tid * Dn + d] - mm) * __expf(-s);
            znew[tid] = zn;
            zsh[tid * Dn + d] = zn;
            lgd -= s;
        }
        __syncthreads();

        // ---- phase D: rank-1 update a_pre += z_d (x) W1m[:,d] ----
        const float wlo = w1col[tid];
        const float whi = w1col[256 + tid];
        #pragma unroll
        for (int r = 0; r < ROWS; ++r) {
            const float zr = znew[r];
            a_pre[r * APAD + tid]       += zr * wlo;
            a_pre[r * APAD + 256 + tid] += zr * whi;
        }
        __syncthreads();
    }

    // ---- outputs: z (B x D) then log_det (B) ----
    for (int i = tid; i < ROWS * Dn; i += NTHR) out[row0 * Dn + i] = zsh[i];
    if (tid < ROWS) out[Bn * Dn + row0 + tid] = lgd;
}

extern "C" void kernel_launch(void* const* d_in, const int* in_sizes, int n_in,
                              void* d_out, int out_size, void* d_ws, size_t ws_size,
                              hipStream_t stream) {
    (void)in_sizes; (void)n_in; (void)out_size; (void)d_ws; (void)ws_size;
    const float* x  = (const float*)d_in[0];
    const float* W1 = (const float*)d_in[1];
    const float* b1 = (const float*)d_in[2];
    const float* W2 = (const float*)d_in[3];
    const float* b2 = (const float*)d_in[4];
    float* out = (float*)d_out;
    iaf_made_inverse<<<dim3(Bn / ROWS), dim3(NTHR), 0, stream>>>(x, W1, b1, W2, b2, out);
}